// GCNModel_1821066133824
// MI455X (gfx1250) — compile-verified
//
#include <hip/hip_runtime.h>

// ---------------------------------------------------------------------------
// GCN (3x GCNConv + BN + ReLU) + 2-layer MLP for MI455X (gfx1250, wave32).
// GEMMs use V_WMMA_F32_16X16X4_F32 (f32 to match the f32 reference).
// W is staged TRANSPOSED in LDS (stride K+2: conflict-free, 8B-aligned) so
// each WMMA step needs one ds_load_b64 for B instead of two strided b32.
// K / NCOL / MODE are template constants -> all addresses fold to immediates.
// Edge aggregation uses float4 gathers + global_atomic_add_f32; every N x 128
// f32 intermediate is 25.6 MB, total workspace ~77 MB < 192 MB L2 (L2-resident).
// ---------------------------------------------------------------------------

typedef __attribute__((ext_vector_type(2))) float v2f;
typedef __attribute__((ext_vector_type(8))) float v8f;

#define BN_EPS 1e-5f

// ---------------------------------------------------------------------------
// Small utility kernels
// ---------------------------------------------------------------------------
__global__ void zero_i32_kernel(int* __restrict__ p, int n) {
    int i = blockIdx.x * blockDim.x + threadIdx.x;
    if (i < n) p[i] = 0;
}

__global__ void fill_f32_kernel(float* __restrict__ p, float v, int n) {
    int i = blockIdx.x * blockDim.x + threadIdx.x;
    if (i < n) p[i] = v;
}

__global__ void count_deg_kernel(const int* __restrict__ dst, int* __restrict__ cnt, int E) {
    int e = blockIdx.x * blockDim.x + threadIdx.x;
    if (e < E) atomicAdd(&cnt[dst[e]], 1);
}

// In-place: read int count, write float deg^{-1/2} (deg = 1 + count, self loop).
__global__ void cnt_to_dinv_kernel(const int* __restrict__ cnt, float* __restrict__ dinv, int n) {
    int i = blockIdx.x * blockDim.x + threadIdx.x;
    if (i < n) {
        int c = cnt[i];
        dinv[i] = rsqrtf(1.0f + (float)c);
    }
}

// ---------------------------------------------------------------------------
// WMMA f32 GEMM:  Out[N x NCOL] = A[N x K] @ W[K x NCOL]  (+ epilogue)
//   MODE 0: plain store (GCN layers: bias folded into the post kernel)
//   MODE 1: + bias, ReLU   (MLP hidden)
//   MODE 2: + bias          (MLP output)
// One 256-thread block = 8 waves; each wave owns a 16x16 output tile.
// LDS holds Wt[NCOL][K+2] (transposed, padded): per k-step B is one
// ds_load_b64 at immediate offset; A is one global b64 load per step.
// ISA f32 16x4 A layout: lanes 0-15 -> K = {k,k+1}, lanes 16-31 -> K = {k+2,k+3};
// B (4x16) mirrors that in K. C/D: VGPR i = row i (lanes 0-15) / row 8+i (16-31).
// ---------------------------------------------------------------------------
template <int K, int NCOL, int MODE>
__global__ void __launch_bounds__(256)
gemm_wmma_f32(const float* __restrict__ A, const float* __restrict__ W,
              const float* __restrict__ bias, float* __restrict__ Out,
              int Nrows) {
    constexpr int LSTR = K + 2;                   // padded LDS stride (even => b64 aligned)
    extern __shared__ float ws[];                 // Wt[NCOL][LSTR]

    // Stage W transposed: global reads coalesced; LDS writes stride LSTR floats
    // (LSTR % 64 == 2 banks step) -> conflict-free across a wave32.
    for (int i = threadIdx.x; i < K * NCOL; i += 256) {
        int r = i / NCOL;                         // k index  (compile-time NCOL => shifts)
        int c = i - r * NCOL;                     // col index
        ws[c * LSTR + r] = W[i];
    }
    __syncthreads();

    const int lane = threadIdx.x & 31;
    const int wave = threadIdx.x >> 5;
    constexpr int tilesPerRow    = NCOL >> 4;     // 8 (NCOL=128) or 4 (NCOL=64)
    constexpr int rowTilesPerBlk = 8 / tilesPerRow;
    const int colTile = wave % tilesPerRow;
    const int rowTile = wave / tilesPerRow;
    const int rowBase = (blockIdx.x * rowTilesPerBlk + rowTile) << 4;
    if (rowBase >= Nrows) return;                 // wave-uniform: EXEC all-1 for WMMA

    const int m     = lane & 15;                  // row-in-tile (A) / col-in-tile (B,C)
    const int khalf = (lane >> 4) << 1;           // 0 for lanes 0-15, 2 for lanes 16-31
    const int col   = (colTile << 4) + m;

    v8f acc = {};
    const float* __restrict__ arow = A + (size_t)(rowBase + m) * K + khalf;
    const float* __restrict__ brow = ws + col * LSTR + khalf;
#pragma unroll
    for (int k = 0; k < K; k += 4) {
        v2f a = *(const v2f*)(arow + k);          // A rows: K,K+1 of this half-wave
        v2f b = *(const v2f*)(brow + k);          // one ds_load_b64, immediate offset
        acc = __builtin_amdgcn_wmma_f32_16x16x4_f32(
                  /*neg_a=*/false, a, /*neg_b=*/false, b,
                  /*c_mod=*/(short)0, acc, /*reuse_a=*/false, /*reuse_b=*/false);
    }

    float bv = 0.0f;
    if (MODE != 0) bv = bias[col];
    const int rshift = (lane >> 4) << 3;          // lanes 16-31 write rows 8..15
#pragma unroll
    for (int i = 0; i < 8; ++i) {
        int row = rowBase + rshift + i;
        float v = acc[i];
        if (MODE != 0) v += bv;
        if (MODE == 1) v = fmaxf(v, 0.0f);
        Out[(size_t)row * NCOL + col] = v;
    }
}

// ---------------------------------------------------------------------------
// Edge scatter:  agg[dst] += h[src] * dinv[src]*dinv[dst]   (F = 128 features)
// One 32-lane wave per edge; each lane handles 4 contiguous features
// (float4 gather from h, 4x global_atomic_add_f32 into agg). All L2-resident.
// ---------------------------------------------------------------------------
__global__ void __launch_bounds__(256)
edge_scatter_kernel(const float* __restrict__ h, const int* __restrict__ src,
                    const int* __restrict__ dst, const float* __restrict__ dinv,
                    float* __restrict__ agg, int E) {
    const int lane = threadIdx.x & 31;
    const int e    = blockIdx.x * 8 + (threadIdx.x >> 5);
    if (e >= E) return;
    const int s = src[e];
    const int d = dst[e];
    const float c = dinv[s] * dinv[d];
    const float4 v = *(const float4*)(h + (size_t)s * 128 + lane * 4);
    float* ap = agg + (size_t)d * 128 + lane * 4;
    atomicAdd(ap + 0, v.x * c);
    atomicAdd(ap + 1, v.y * c);
    atomicAdd(ap + 2, v.z * c);
    atomicAdd(ap + 3, v.w * c);
}

// ---------------------------------------------------------------------------
// Fused epilogue per GCN layer:
//   out = ReLU( BN( agg + h*dinv^2 + b ) )   (eval-mode BN; dropout = identity)
// ---------------------------------------------------------------------------
__global__ void __launch_bounds__(256)
post_bn_relu_kernel(const float* __restrict__ agg, const float* __restrict__ h,
                    const float* __restrict__ dinv, const float* __restrict__ b,
                    const float* __restrict__ g, const float* __restrict__ be,
                    const float* __restrict__ mu, const float* __restrict__ var,
                    float* __restrict__ out, int total) {
    int i = blockIdx.x * blockDim.x + threadIdx.x;
    if (i >= total) return;
    const int node = i >> 7;
    const int f    = i & 127;
    const float di  = dinv[node];
    const float val = agg[i] + h[i] * (di * di) + b[f];
    const float sc  = g[f] * rsqrtf(var[f] + BN_EPS);
    out[i] = fmaxf((val - mu[f]) * sc + be[f], 0.0f);
}

// ---------------------------------------------------------------------------
// Launcher
// ---------------------------------------------------------------------------
extern "C" void kernel_launch(void* const* d_in, const int* in_sizes, int n_in,
                              void* d_out, int out_size, void* d_ws, size_t ws_size,
                              hipStream_t stream) {
    const float* x   = (const float*)d_in[0];
    const int*   src = (const int*)d_in[1];
    const int*   dst = (const int*)d_in[2];
    const float* Wl[3]  = {(const float*)d_in[3],  (const float*)d_in[9],  (const float*)d_in[15]};
    const float* bl[3]  = {(const float*)d_in[4],  (const float*)d_in[10], (const float*)d_in[16]};
    const float* gl[3]  = {(const float*)d_in[5],  (const float*)d_in[11], (const float*)d_in[17]};
    const float* bel[3] = {(const float*)d_in[6],  (const float*)d_in[12], (const float*)d_in[18]};
    const float* ml[3]  = {(const float*)d_in[7],  (const float*)d_in[13], (const float*)d_in[19]};
    const float* vl[3]  = {(const float*)d_in[8],  (const float*)d_in[14], (const float*)d_in[20]};
    const float* Wm1 = (const float*)d_in[21];
    const float* bm1 = (const float*)d_in[22];
    const float* Wm2 = (const float*)d_in[23];
    const float* bm2 = (const float*)d_in[24];

    const int N = in_sizes[0] / 64;   // 50000 (multiple of 16)
    const int E = in_sizes[1];        // 800000
    const int HID = 128;
    const int NF = N * HID;           // 6.4M

    // Workspace carve-out (~77 MB): dinv | h | agg | act
    char* wsp = (char*)d_ws;
    size_t off = 0;
    auto carve = [&](size_t bytes) -> void* {
        void* p = (void*)(wsp + off);
        off = (off + bytes + 255) & ~(size_t)255;
        return p;
    };
    float* dinv = (float*)carve((size_t)N * 4);
    float* hbuf = (float*)carve((size_t)NF * 4);
    float* agg  = (float*)carve((size_t)NF * 4);
    float* act  = (float*)carve((size_t)NF * 4);
    (void)ws_size; (void)n_in; (void)out_size;

    // --- degrees: count into dinv region (as int), convert in place ---
    zero_i32_kernel<<<(N + 255) / 256, 256, 0, stream>>>((int*)dinv, N);
    count_deg_kernel<<<(E + 255) / 256, 256, 0, stream>>>(dst, (int*)dinv, E);
    cnt_to_dinv_kernel<<<(N + 255) / 256, 256, 0, stream>>>((const int*)dinv, dinv, N);

    // LDS bytes for transposed-W staging: NCOL * (K+2) * 4
    const size_t lds64  = (size_t)128 * (64 + 2) * 4;    // K=64,  NCOL=128
    const size_t lds128 = (size_t)128 * (128 + 2) * 4;   // K=128, NCOL=128
    const size_t lds64c = (size_t)64  * (128 + 2) * 4;   // K=128, NCOL=64

    // --- 3 GCN layers ---
    const float* hin = x;
    for (int l = 0; l < 3; ++l) {
        if (l == 0) {
            gemm_wmma_f32<64, 128, 0><<<(N + 15) / 16, 256, lds64, stream>>>(
                hin, Wl[l], nullptr, hbuf, N);
        } else {
            gemm_wmma_f32<128, 128, 0><<<(N + 15) / 16, 256, lds128, stream>>>(
                hin, Wl[l], nullptr, hbuf, N);
        }
        fill_f32_kernel<<<(NF + 255) / 256, 256, 0, stream>>>(agg, 0.0f, NF);
        edge_scatter_kernel<<<(E + 7) / 8, 256, 0, stream>>>(hbuf, src, dst, dinv, agg, E);
        post_bn_relu_kernel<<<(NF + 255) / 256, 256, 0, stream>>>(
            agg, hbuf, dinv, bl[l], gl[l], bel[l], ml[l], vl[l], act, NF);
        hin = act;
    }

    // --- MLP head ---
    gemm_wmma_f32<128, 128, 1><<<(N + 15) / 16, 256, lds128, stream>>>(
        act, Wm1, bm1, hbuf, N);                              // bias + ReLU
    gemm_wmma_f32<128, 64, 2><<<(N + 31) / 32, 256, lds64c, stream>>>(
        hbuf, Wm2, bm2, (float*)d_out, N);                    // bias
}